// ScaledDotProductAttention_10333691314182
// MI455X (gfx1250) — compile-verified
//
#include <hip/hip_runtime.h>
#include <hip/hip_bf16.h>

// Scaled dot-product attention with inner_dis scaling + mask, returning
// (output[B,H,S,D], attn[B,H,S,S]) — fully fused per 16-row query strip.
// CDNA5 (gfx1250): wave32, V_WMMA_F32_16X16X4_F32, 154KB LDS-resident strip.
// Round 2: dual-chain WMMA accumulation for ILP, 64-row V staging blocks
// (half the barriers, 8 independent PV chains/wave), K-tile prefetch.

#define B_  4
#define H_  8
#define S_  2048
#define D_  64
#define NDIS 8
#define INVT 0.125f        // 1/TEMPERATURE

#define PSTR 2052          // padded LDS row stride for score strip (floats)
#define VSTR 68            // padded LDS row stride for V tile (floats, 16B-aligned rows)
#define OSTR 68            // padded LDS row stride for O accumulator
#define VBLK 64            // V staging block rows

typedef float v2f __attribute__((ext_vector_type(2)));
typedef float v8f __attribute__((ext_vector_type(8)));

__launch_bounds__(256)
__global__ void sdpa_fused_kernel(const float* __restrict__ q,
                                  const float* __restrict__ k,
                                  const float* __restrict__ v,
                                  const float* __restrict__ dis,
                                  const unsigned char* __restrict__ mask,
                                  float* __restrict__ out,
                                  float* __restrict__ attn)
{
    __shared__ float sP[16 * PSTR];    // 131328 B: e-values strip (16 rows x 2048)
    __shared__ float sV[VBLK * VSTR];  //  17408 B: staged V block
    __shared__ float sO[16 * OSTR];    //   4352 B: cross-wave O accumulator
    __shared__ float sRed[16 * 16];    //   1024 B: row reductions
    __shared__ float sRowMax[16];
    __shared__ float sRowSum[16];

    const int tid   = threadIdx.x;
    const int wave  = tid >> 5;            // 8 waves of 32
    const int lane  = tid & 31;
    const int ln    = lane & 15;           // N (or M) index within half-wave
    const int khalf = (lane < 16) ? 0 : 2; // K pair selector per WMMA f32 layout
    const int mbase = (lane < 16) ? 0 : 8; // C/D row-half selector

    const int rb = blockIdx.x * 16;        // query-row base of this strip
    const int h  = blockIdx.y;
    const int b  = blockIdx.z;

    const float* qp = q + (((size_t)b * H_ + h) * S_ + rb) * D_;
    const float* kp = k + (((size_t)b * H_ + h) * S_) * D_;
    const float* vp = v + (((size_t)b * H_ + h) * S_) * D_;
    const float* dp = dis + ((size_t)b * S_ + rb) * S_;
    const unsigned char* mp = mask + ((size_t)b * S_ + rb) * S_;
    float* op = out  + (((size_t)b * H_ + h) * S_ + rb) * D_;
    float* ap = attn + (((size_t)b * H_ + h) * S_ + rb) * S_;

    // ---------------- Phase 1: raw scores (Q/T)·K^T into LDS ----------------
    // A fragment (Q 16x4): lanes 0-15 -> M=0..15 K={4s,4s+1}; lanes 16-31 K={4s+2,4s+3}
    v2f qa[16];
#pragma unroll
    for (int s = 0; s < 16; ++s) {
        qa[s] = *reinterpret_cast<const v2f*>(qp + (size_t)ln * D_ + 4 * s + khalf);
    }

    for (int ct = wave; ct < S_ / 16; ct += 8) {
        const int tb = ct * 16;
        // prefetch this wave's next K tile while the matrix pipe is busy
        if (tb + 128 < S_) {
            __builtin_prefetch(kp + (size_t)(tb + 128 + ln) * D_, 0, 1);
        }
        // B fragment (K^T 4x16): B[kk][n] = K[tb+n][4s+kk]
        v2f kb[16];
#pragma unroll
        for (int s = 0; s < 16; ++s) {
            kb[s] = *reinterpret_cast<const v2f*>(kp + (size_t)(tb + ln) * D_ + 4 * s + khalf);
        }
        // two independent accumulation chains for ILP on the XDL pipe
        v8f acc0 = {}, acc1 = {};
#pragma unroll
        for (int s = 0; s < 16; s += 2) {
            acc0 = __builtin_amdgcn_wmma_f32_16x16x4_f32(
                false, qa[s],     false, kb[s],     (short)0, acc0, false, false);
            acc1 = __builtin_amdgcn_wmma_f32_16x16x4_f32(
                false, qa[s + 1], false, kb[s + 1], (short)0, acc1, false, false);
        }
        const v8f acc = acc0 + acc1;
        // C layout: VGPR j -> (M = mbase + j, N = ln); store scaled by 1/T
#pragma unroll
        for (int j = 0; j < 8; ++j) {
            sP[(mbase + j) * PSTR + tb + ln] = acc[j] * INVT;
        }
    }

    // zero O accumulator while scores settle
    for (int i = tid; i < 16 * OSTR; i += 256) sO[i] = 0.0f;
    __syncthreads();

    // 16 threads per row: row = tid>>4, this thread's column slot = tid&15
    const int srow = tid >> 4;
    const int sl   = tid & 15;
    float* prow = &sP[srow * PSTR];

    // ---------------- Phase 2: inner_dis scale + mask + row max -------------
    {
        const float* drow = dp + (size_t)srow * S_;
        const unsigned char* mrow = mp + (size_t)srow * S_;
        const bool usedis = (h < NDIS);
        float m = -3.0e38f;
#pragma unroll 4
        for (int j = 0; j < 32; ++j) {
            const int c = 4 * (sl + 16 * j);
            float4 x = *reinterpret_cast<float4*>(prow + c);
            if (usedis) {
                float4 dd = *reinterpret_cast<const float4*>(drow + c);
                x.x *= dd.x; x.y *= dd.y; x.z *= dd.z; x.w *= dd.w;
            }
            const unsigned int mm = *reinterpret_cast<const unsigned int*>(mrow + c);
            if (mm & 0x000000FFu) x.x = -1e9f;
            if (mm & 0x0000FF00u) x.y = -1e9f;
            if (mm & 0x00FF0000u) x.z = -1e9f;
            if (mm & 0xFF000000u) x.w = -1e9f;
            *reinterpret_cast<float4*>(prow + c) = x;
            m = fmaxf(m, fmaxf(fmaxf(x.x, x.y), fmaxf(x.z, x.w)));
        }
        sRed[srow * 16 + sl] = m;
    }
    __syncthreads();
    if (tid < 16) {
        float m = sRed[tid * 16];
#pragma unroll
        for (int l = 1; l < 16; ++l) m = fmaxf(m, sRed[tid * 16 + l]);
        sRowMax[tid] = m;
    }
    __syncthreads();

    // ---------------- Phase 3: exp(x - rowmax) + row sum --------------------
    {
        const float rm = sRowMax[srow];
        float ssum = 0.0f;
#pragma unroll 4
        for (int j = 0; j < 32; ++j) {
            const int c = 4 * (sl + 16 * j);
            float4 x = *reinterpret_cast<float4*>(prow + c);
            x.x = __expf(x.x - rm);
            x.y = __expf(x.y - rm);
            x.z = __expf(x.z - rm);
            x.w = __expf(x.w - rm);
            *reinterpret_cast<float4*>(prow + c) = x;
            ssum += (x.x + x.y) + (x.z + x.w);
        }
        sRed[srow * 16 + sl] = ssum;
    }
    __syncthreads();
    if (tid < 16) {
        float ssum = 0.0f;
#pragma unroll
        for (int l = 0; l < 16; ++l) ssum += sRed[tid * 16 + l];
        sRowSum[tid] = ssum;
    }
    __syncthreads();

    // ---------------- Phase 4: stream normalized attn to global -------------
    {
        const float inv = 1.0f / sRowSum[srow];
        float* arow = ap + (size_t)srow * S_;
#pragma unroll 4
        for (int j = 0; j < 32; ++j) {
            const int c = 4 * (sl + 16 * j);
            float4 x = *reinterpret_cast<float4*>(prow + c);
            x.x *= inv; x.y *= inv; x.z *= inv; x.w *= inv;
            *reinterpret_cast<float4*>(arow + c) = x;  // coalesced b128 stores
        }
    }

    // ---------------- Phase 5: O = P·V (unnormalized P, divide at end) ------
    // 64-row V blocks; each wave owns K-steps t0 = 4*wave and t0 = 4*wave+32,
    // with a separate accumulator chain per (k-half, d-tile): 8 independent chains.
    v8f oa0 = {}, oa1 = {}, oa2 = {}, oa3 = {};
    v8f ob0 = {}, ob1 = {}, ob2 = {}, ob3 = {};
    for (int tb = 0; tb < S_; tb += VBLK) {
        __syncthreads();
        // cooperative stage of V[tb..tb+64) x 64 into LDS (4 float4 per thread)
#pragma unroll
        for (int j2 = 0; j2 < 4; ++j2) {
            const int f  = tid + 256 * j2;      // float4 index 0..1023
            const int r  = f >> 4;
            const int cb = (f & 15) * 4;
            float4 x = *reinterpret_cast<const float4*>(vp + (size_t)(tb + r) * D_ + cb);
            *reinterpret_cast<float4*>(&sV[r * VSTR + cb]) = x;
        }
        __syncthreads();
        const int tA = 4 * wave;        // local K-step A
        const int tB = 4 * wave + 32;   // local K-step B
        const v2f paA = *reinterpret_cast<const v2f*>(&sP[ln * PSTR + tb + tA + khalf]);
        const v2f paB = *reinterpret_cast<const v2f*>(&sP[ln * PSTR + tb + tB + khalf]);
        v2f vb;
        vb.x = sV[(tA + khalf) * VSTR + 0 * 16 + ln];
        vb.y = sV[(tA + khalf + 1) * VSTR + 0 * 16 + ln];
        oa0 = __builtin_amdgcn_wmma_f32_16x16x4_f32(false, paA, false, vb, (short)0, oa0, false, false);
        vb.x = sV[(tA + khalf) * VSTR + 1 * 16 + ln];
        vb.y = sV[(tA + khalf + 1) * VSTR + 1 * 16 + ln];
        oa1 = __builtin_amdgcn_wmma_f32_16x16x4_f32(false, paA, false, vb, (short)0, oa1, false, false);
        vb.x = sV[(tA + khalf) * VSTR + 2 * 16 + ln];
        vb.y = sV[(tA + khalf + 1) * VSTR + 2 * 16 + ln];
        oa2 = __builtin_amdgcn_wmma_f32_16x16x4_f32(false, paA, false, vb, (short)0, oa2, false, false);
        vb.x = sV[(tA + khalf) * VSTR + 3 * 16 + ln];
        vb.y = sV[(tA + khalf + 1) * VSTR + 3 * 16 + ln];
        oa3 = __builtin_amdgcn_wmma_f32_16x16x4_f32(false, paA, false, vb, (short)0, oa3, false, false);

        vb.x = sV[(tB + khalf) * VSTR + 0 * 16 + ln];
        vb.y = sV[(tB + khalf + 1) * VSTR + 0 * 16 + ln];
        ob0 = __builtin_amdgcn_wmma_f32_16x16x4_f32(false, paB, false, vb, (short)0, ob0, false, false);
        vb.x = sV[(tB + khalf) * VSTR + 1 * 16 + ln];
        vb.y = sV[(tB + khalf + 1) * VSTR + 1 * 16 + ln];
        ob1 = __builtin_amdgcn_wmma_f32_16x16x4_f32(false, paB, false, vb, (short)0, ob1, false, false);
        vb.x = sV[(tB + khalf) * VSTR + 2 * 16 + ln];
        vb.y = sV[(tB + khalf + 1) * VSTR + 2 * 16 + ln];
        ob2 = __builtin_amdgcn_wmma_f32_16x16x4_f32(false, paB, false, vb, (short)0, ob2, false, false);
        vb.x = sV[(tB + khalf) * VSTR + 3 * 16 + ln];
        vb.y = sV[(tB + khalf + 1) * VSTR + 3 * 16 + ln];
        ob3 = __builtin_amdgcn_wmma_f32_16x16x4_f32(false, paB, false, vb, (short)0, ob3, false, false);
    }

    // merge the two K-halves, then cross-wave reduction via LDS float atomics
    const v8f o0 = oa0 + ob0;
    const v8f o1 = oa1 + ob1;
    const v8f o2 = oa2 + ob2;
    const v8f o3 = oa3 + ob3;
#pragma unroll
    for (int j = 0; j < 8; ++j) {
        atomicAdd(&sO[(mbase + j) * OSTR + 0 * 16 + ln], o0[j]);
        atomicAdd(&sO[(mbase + j) * OSTR + 1 * 16 + ln], o1[j]);
        atomicAdd(&sO[(mbase + j) * OSTR + 2 * 16 + ln], o2[j]);
        atomicAdd(&sO[(mbase + j) * OSTR + 3 * 16 + ln], o3[j]);
    }
    __syncthreads();

    // ---------------- Phase 6: normalize + write output ---------------------
    {
        const float inv = 1.0f / sRowSum[srow];
        float4 x = *reinterpret_cast<float4*>(&sO[srow * OSTR + sl * 4]);
        x.x *= inv; x.y *= inv; x.z *= inv; x.w *= inv;
        *reinterpret_cast<float4*>(op + (size_t)srow * D_ + sl * 4) = x;
    }
}

extern "C" void kernel_launch(void* const* d_in, const int* in_sizes, int n_in,
                              void* d_out, int out_size, void* d_ws, size_t ws_size,
                              hipStream_t stream) {
    const float* q   = (const float*)d_in[0];
    const float* k   = (const float*)d_in[1];
    const float* v   = (const float*)d_in[2];
    const float* dis = (const float*)d_in[3];
    const unsigned char* mask = (const unsigned char*)d_in[4];

    float* out  = (float*)d_out;
    float* attn = out + (size_t)B_ * H_ * S_ * D_;  // tuple order: (output, attn)

    dim3 grid(S_ / 16, H_, B_);
    sdpa_fused_kernel<<<grid, 256, 0, stream>>>(q, k, v, dis, mask, out, attn);
}